// Grid1_78993038508382
// MI455X (gfx1250) — compile-verified
//
#include <hip/hip_runtime.h>

// Problem constants (match the reference file's globals).
#define GC     4
#define GRID_H 1024
#define GRID_W 1024
#define OUT_H  4096
#define OUT_W  4096

// Tile geometry: one block = 256 threads covering 8 output rows x 256 cols.
#define TROWS 8
#define TCOLS 256
// Staged grid band: guard row/col on the low side (fp32 floor() can land one
// below the nominal integer coordinate) and +1 on the high side for x1/y1.
#define LROWS (TROWS + 2)              // 10 grid rows
#define LCOLS (TCOLS + 2)              // 258 grid cols
#define LCH   (LROWS * LCOLS)          // 2580 floats per channel
#define LTOT  (GC * LCH)               // 10320 floats = 41280 B LDS

// ---- CDNA5 async global->LDS staging (ASYNCcnt path) ------------------------
#if defined(__has_builtin)
#if __has_builtin(__builtin_amdgcn_global_load_async_to_lds_b32)
#define HAVE_ASYNC_LDS 1
#endif
#endif
#ifndef HAVE_ASYNC_LDS
#define HAVE_ASYNC_LDS 0
#endif

typedef __attribute__((address_space(1))) int g_i32;   // global (device) int
typedef __attribute__((address_space(3))) int l_i32;   // LDS int

__device__ __forceinline__ void stage_word(const float* gsrc, float* ldst) {
#if HAVE_ASYNC_LDS
    // Per-lane gather: copies the word at global address gsrc to LDS address ldst; tracked with ASYNCcnt.
    __builtin_amdgcn_global_load_async_to_lds_b32(
        (g_i32*)gsrc, (l_i32*)ldst, 0, 0);
#else
    *ldst = *gsrc;                      // fallback: sync load + ds_store
#endif
}

__device__ __forceinline__ void stage_drain() {
#if HAVE_ASYNC_LDS
#if __has_builtin(__builtin_amdgcn_s_wait_asynccnt)
    __builtin_amdgcn_s_wait_asynccnt(0);
#else
    asm volatile("s_wait_asynccnt 0x0" ::: "memory");
#endif
#endif
}
// -----------------------------------------------------------------------------

__global__ __launch_bounds__(256) void bilinear_tiled_kernel(
    const float* __restrict__ grid,     // [GC][GRID_H][GRID_W]
    const int*   __restrict__ cstart,   // {sx, sy}
    float*       __restrict__ out)      // [GC][OUT_H][OUT_W]
{
    __shared__ float tile[LTOT];

    const int tid = threadIdx.x;
    const int ox0 = blockIdx.x * TCOLS;
    const int oy0 = blockIdx.y * TROWS;
    const int sx  = cstart[0];
    const int sy  = cstart[1];

    const int vxbase = (sx + ox0) & (GRID_W - 1);
    const int vybase = (sy + oy0) & (GRID_H - 1);

    // ---- Stage the wrapped (LROWS x LCOLS) grid band, all channels ----------
    // Per-lane addressing absorbs the %1024 wrap in both axes.
    const int gcol0 = (vxbase - 1 + tid) & (GRID_W - 1);
    const int gcolT = (vxbase - 1 + TCOLS + tid) & (GRID_W - 1);

    for (int c = 0; c < GC; ++c) {
        const float* gch = grid + (size_t)c * ((size_t)GRID_H * GRID_W);
        for (int r = 0; r < LROWS; ++r) {
            const int grow = (vybase - 1 + r) & (GRID_H - 1);
            const float* grp = gch + (size_t)grow * GRID_W;
            float* lrp = &tile[(c * LROWS + r) * LCOLS];
            stage_word(grp + gcol0, lrp + tid);             // cols 0..255
            if (tid < 2) {
                stage_word(grp + gcolT, lrp + TCOLS + tid); // cols 256,257
            }
        }
    }
    stage_drain();        // s_wait_asynccnt 0: my copies landed in LDS
    __syncthreads();      // everyone's copies visible

    // ---- x coordinate chain: once per thread, reused across 8 rows ----------
    const float offs     = 511.5f;
    const float inv_offs = 1.0f / 511.5f;
    const float maxc     = 1023.0f;

    const int ox = ox0 + tid;
    const int vx = (sx + ox) & (GRID_W - 1);
    float cx = ((float)vx - offs) * inv_offs;
    float x  = (cx + 1.0f) * 0.5f * maxc;
    x = fminf(fmaxf(x, 0.0f), maxc);
    const float x0f  = floorf(x);
    const float wx   = x - x0f;
    const float omwx = 1.0f - wx;
    const int   x0   = (int)x0f;
    const int   x1   = min(x0 + 1, GRID_W - 1);
    const bool  xclip = (x1 == x0);                      // border clip at col 1023
    const int   t0   = (x0 - vxbase + 1) & (GRID_W - 1); // tile col: tid or tid+1

    const size_t outPlane = (size_t)OUT_H * OUT_W;

    for (int i = 0; i < TROWS; ++i) {
        const int oy = oy0 + i;

        // y chain is block-uniform -> scalarizes to SALU float ops.
        const int vy = (sy + oy) & (GRID_H - 1);
        float cy = ((float)vy - offs) * inv_offs;
        float y  = (cy + 1.0f) * 0.5f * maxc;
        y = fminf(fmaxf(y, 0.0f), maxc);
        const float y0f  = floorf(y);
        const float wy   = y - y0f;
        const float omwy = 1.0f - wy;
        const int   y0   = (int)y0f;
        const int   y1   = min(y0 + 1, GRID_H - 1);      // border clip at row 1023
        const int   r0   = (y0 - vybase + 1) & (GRID_H - 1);
        const int   r1   = (y1 - vybase + 1) & (GRID_H - 1);

        const int lr0 = r0 * LCOLS + t0;
        const int lr1 = r1 * LCOLS + t0;
        const size_t outIdx = (size_t)oy * OUT_W + ox;

#pragma unroll
        for (int c = 0; c < GC; ++c) {
            const float* lt = &tile[c * LCH];
            // Adjacent-pair reads -> ds_load_2addr_b32 fusion.
            const float v00  = lt[lr0];
            const float v01p = lt[lr0 + 1];
            const float v10  = lt[lr1];
            const float v11p = lt[lr1 + 1];
            const float v01 = xclip ? v00 : v01p;        // x1==x0 border case
            const float v11 = xclip ? v10 : v11p;

            const float top = v00 * omwx + v01 * wx;     // reference association
            const float bot = v10 * omwx + v11 * wx;
            const float res = top * omwy + bot * wy;

            // 256MB streaming output: NT store, bypass cache retention.
            __builtin_nontemporal_store(res, out + (size_t)c * outPlane + outIdx);
        }
    }
}

extern "C" void kernel_launch(void* const* d_in, const int* in_sizes, int n_in,
                              void* d_out, int out_size, void* d_ws, size_t ws_size,
                              hipStream_t stream) {
    (void)in_sizes; (void)n_in; (void)out_size; (void)d_ws; (void)ws_size;

    const float* grid   = (const float*)d_in[0];   // (1, 4, 1024, 1024) fp32
    const int*   cstart = (const int*)d_in[1];     // (2,) int32 {sx, sy}
    float*       out    = (float*)d_out;           // (1, 4, 4096, 4096) fp32

    dim3 block(TCOLS);                              // 8 wave32 waves
    dim3 gridDim(OUT_W / TCOLS, OUT_H / TROWS);     // (16, 512) blocks
    bilinear_tiled_kernel<<<gridDim, block, 0, stream>>>(grid, cstart, out);
}